// NerfNet_57415122813506
// MI455X (gfx1250) — compile-verified
//
#include <hip/hip_runtime.h>
#include <math.h>

// ---------------------------------------------------------------------------
// NeRF-style volume rendering for MI455X (gfx1250, wave32).
//
// Per ray (N=128 samples):
//   d_i   = |ray_d| * (z_{i+1}-z_i)            (last closed by z_max)
//   t_i   = exp(-sigma_i d_i) + 1e-6
//   Texcl = cumprod_strict(t) = exp(cumsum_strict(log t))   <- WMMA scan
//   w_i   = (1 + 1e-6 - t_i) * Texcl_i
//   ldist = 2 * sum_i w_i (m_i * Wexcl_i - Sexcl_i) + (1/3) sum w_i^2 d_i
//           (exact rewrite of sum_{ij} w_i w_j |m_i - m_j| for sorted m)
// The three exclusive prefix sums are computed as [16x128] x [128x128]
// matmuls against a strict-upper-triangular ones matrix using
// V_WMMA_F32_16X16X4_F32 (B operand synthesized from lane indices).
// ---------------------------------------------------------------------------

typedef float v2f __attribute__((ext_vector_type(2)));
typedef float v8f __attribute__((ext_vector_type(8)));

#define TINY  1e-6f
#define NS    128          // samples per ray
#define RPB   16           // rays per block (one wave32 per block)
#define SR    132          // padded LDS row stride in floats

// Strict exclusive prefix-sum over the sample axis for a 16x128 LDS tile:
//   out[r][i] = sum_{j<i} in[r][j]
// Implemented as Out = In * U with U[j][i] = (j < i), via 16x16x4 f32 WMMA.
__device__ __forceinline__ void scan_strict(const float* __restrict__ in,
                                            float* __restrict__ out,
                                            int lane) {
  const int row = lane & 15;     // A-matrix row (ray)
  const int hi  = lane >> 4;     // 0: K=0/1, 1: K=2/3  (ISA f32 A layout)
  const int col = lane & 15;     // B/D column
  const int k2  = 2 * hi;
  for (int n = 0; n < 8; ++n) {  // output column tiles (16 cols each)
    v8f acc = {0.f, 0.f, 0.f, 0.f, 0.f, 0.f, 0.f, 0.f};
    const int nb = 16 * n + col;
#pragma unroll
    for (int c = 0; c < 32; ++c) {     // K chunks of 4
      const float* ap = in + row * SR + 4 * c + k2;
      v2f a;
      a.x = ap[0];                      // A[row][4c+k2]
      a.y = ap[1];                      // A[row][4c+k2+1]
      v2f b;                            // U[j][i] = (j < i)
      b.x = ((4 * c + k2)     < nb) ? 1.0f : 0.0f;
      b.y = ((4 * c + k2 + 1) < nb) ? 1.0f : 0.0f;
      acc = __builtin_amdgcn_wmma_f32_16x16x4_f32(
          false, a, false, b, (short)0, acc, false, false);
    }
#pragma unroll
    for (int j = 0; j < 8; ++j)         // D layout: VGPR j -> row j (+8 hi)
      out[(j + 8 * hi) * SR + 16 * n + col] = acc[j];
  }
}

__global__ __launch_bounds__(32) void nerf_render_kernel(
    const float* __restrict__ ray_d,      // [R,3]
    const float* __restrict__ fg_z_max,   // [R]
    const float* __restrict__ fg_z_vals,  // [R,N]
    const float* __restrict__ sigma,      // [R,N]
    const float* __restrict__ rgb,        // [R,N,3]
    const float* __restrict__ bg_rgb,     // [R,3]
    float* __restrict__ o_rgbmap,         // [R,3]
    float* __restrict__ o_w,              // [R,N]
    float* __restrict__ o_fgrgb,          // [R,3]
    float* __restrict__ o_depth,          // [R]
    float* __restrict__ o_bg,             // [R]
    float* __restrict__ o_ld,             // [R]
    float* __restrict__ o_tv) {           // [R]

  __shared__ __align__(16) float sZ[RPB * SR];   // z_vals
  __shared__ __align__(16) float sD[RPB * SR];   // dists
  __shared__ __align__(16) float sM[RPB * SR];   // scaled midpoints
  __shared__ __align__(16) float sW[RPB * SR];   // weights
  __shared__ __align__(16) float sA[RPB * SR];   // log t   -> w*m
  __shared__ __align__(16) float sB[RPB * SR];   // cumsum(log t) -> Wexcl
  __shared__ __align__(16) float sC[RPB * SR];   // t       -> Sexcl
  __shared__ float sNorm[RPB];
  __shared__ float sZmax[RPB];
  __shared__ float sBgl[RPB];

  const int lane = threadIdx.x;
  const int ray0 = blockIdx.x * RPB;

  // ---- per-ray scalars ----
  if (lane < RPB) {
    const int g = ray0 + lane;
    const float dx = ray_d[3 * g], dy = ray_d[3 * g + 1], dz = ray_d[3 * g + 2];
    sNorm[lane] = sqrtf(dx * dx + dy * dy + dz * dz);
    sZmax[lane] = fg_z_max[g];
  }
  // ---- stage z_vals (contiguous 16*128 floats) ----
  for (int k = 0; k < 64; ++k) {
    const int e = lane + 32 * k;
    sZ[(e >> 7) * SR + (e & 127)] = fg_z_vals[(size_t)ray0 * NS + e];
  }
  __syncthreads();

  // ---- dists, t, log t, midpoints ----
  for (int k = 0; k < 64; ++k) {
    const int e = lane + 32 * k;
    const int r = e >> 7, i = e & 127;
    const float z   = sZ[r * SR + i];
    const float zn  = (i < NS - 1) ? sZ[r * SR + i + 1] : sZmax[r];
    const float nrm = sNorm[r];
    const float d   = nrm * (zn - z);
    const float t   = expf(-sigma[(size_t)ray0 * NS + e] * d) + TINY;
    sD[r * SR + i] = d;
    sC[r * SR + i] = t;
    sA[r * SR + i] = logf(t);
    sM[r * SR + i] = 0.5f * nrm * (z + zn);
  }
  __syncthreads();

  // ---- transmittance scan (WMMA): sB = cumsum_strict(log t) ----
  scan_strict(sA, sB, lane);
  __syncthreads();

  // ---- weights, bg_lambda, P = w*m ----
  for (int k = 0; k < 64; ++k) {
    const int e = lane + 32 * k;
    const int r = e >> 7, i = e & 127;
    const float t     = sC[r * SR + i];
    const float Texcl = expf(sB[r * SR + i]);
    const float w     = (1.0f + TINY - t) * Texcl;   // alpha * Texcl
    sW[r * SR + i] = w;
    if (i == NS - 1) sBgl[r] = Texcl * t;            // T inclusive at N-1
    sA[r * SR + i] = w * sM[r * SR + i];             // reuse sA for w*m
  }
  __syncthreads();

  // ---- exclusive sums for the distortion term (WMMA scans) ----
  scan_strict(sW, sB, lane);   // sB = Wexcl
  __syncthreads();
  scan_strict(sA, sC, lane);   // sC = Sexcl
  __syncthreads();

  // ---- per-ray reductions ----
  for (int r = 0; r < RPB; ++r) {
    float s0 = 0.f, s1 = 0.f, s2 = 0.f, sd = 0.f, sl = 0.f, st = 0.f;
    const float* rrow = rgb + (size_t)(ray0 + r) * (NS * 3);
#pragma unroll
    for (int k = 0; k < 4; ++k) {
      const int i = lane + 32 * k;
      const float w  = sW[r * SR + i];
      const float z  = sZ[r * SR + i];
      const float m  = sM[r * SR + i];
      const float we = sB[r * SR + i];
      const float se = sC[r * SR + i];
      const float d  = sD[r * SR + i];
      sd += w * z;
      sl += 2.0f * w * (m * we - se) + (1.0f / 3.0f) * (w * w) * d;
      if (i > 0) st += fabsf(w - sW[r * SR + i - 1]);
      s0 += w * rrow[3 * i + 0];
      s1 += w * rrow[3 * i + 1];
      s2 += w * rrow[3 * i + 2];
    }
#pragma unroll
    for (int off = 16; off > 0; off >>= 1) {
      s0 += __shfl_xor(s0, off, 32);
      s1 += __shfl_xor(s1, off, 32);
      s2 += __shfl_xor(s2, off, 32);
      sd += __shfl_xor(sd, off, 32);
      sl += __shfl_xor(sl, off, 32);
      st += __shfl_xor(st, off, 32);
    }
    if (lane == 0) {
      const int g = ray0 + r;
      const float bl = sBgl[r];
      o_fgrgb[3 * g + 0] = s0;
      o_fgrgb[3 * g + 1] = s1;
      o_fgrgb[3 * g + 2] = s2;
      o_rgbmap[3 * g + 0] = s0 + bl * bg_rgb[3 * g + 0];
      o_rgbmap[3 * g + 1] = s1 + bl * bg_rgb[3 * g + 1];
      o_rgbmap[3 * g + 2] = s2 + bl * bg_rgb[3 * g + 2];
      o_depth[g] = sd;
      o_bg[g]    = bl;
      o_ld[g]    = sl;
      o_tv[g]    = st;
    }
  }

  // ---- write weights (contiguous) ----
  for (int k = 0; k < 64; ++k) {
    const int e = lane + 32 * k;
    o_w[(size_t)ray0 * NS + e] = sW[(e >> 7) * SR + (e & 127)];
  }
}

extern "C" void kernel_launch(void* const* d_in, const int* in_sizes, int n_in,
                              void* d_out, int out_size, void* d_ws, size_t ws_size,
                              hipStream_t stream) {
  const float* ray_d   = (const float*)d_in[0];
  const float* z_max   = (const float*)d_in[1];
  const float* z_vals  = (const float*)d_in[2];
  const float* sigma   = (const float*)d_in[3];
  const float* rgb     = (const float*)d_in[4];
  const float* bg_rgb  = (const float*)d_in[5];

  const int R = in_sizes[1];          // fg_z_max has R elements

  float* out = (float*)d_out;
  // output tuple, flattened in return order
  float* o_rgbmap = out;              // [R,3]
  float* o_w      = o_rgbmap + (size_t)R * 3;   // [R,128]
  float* o_fgrgb  = o_w + (size_t)R * NS;       // [R,3]
  float* o_depth  = o_fgrgb + (size_t)R * 3;    // [R]
  float* o_bg     = o_depth + R;                // [R]
  float* o_ld     = o_bg + R;                   // [R]
  float* o_tv     = o_ld + R;                   // [R]

  dim3 grid(R / RPB), block(32);
  nerf_render_kernel<<<grid, block, 0, stream>>>(
      ray_d, z_max, z_vals, sigma, rgb, bg_rgb,
      o_rgbmap, o_w, o_fgrgb, o_depth, o_bg, o_ld, o_tv);
}